// SPDNetwork_3_53240414601664
// MI455X (gfx1250) — compile-verified
//
#include <hip/hip_runtime.h>

typedef float v2f __attribute__((ext_vector_type(2)));
typedef float v4f __attribute__((ext_vector_type(4)));
typedef float v8f __attribute__((ext_vector_type(8)));

#define REC_EPS   1e-4f
#define LOG_FLOOR 1e-10f

// One branchless Jacobi rotation on pair (p,q); all indices compile-time constants.
#define JROT(p, q)                                                                 \
  do {                                                                             \
    float apq = a[p][q];                                                           \
    float dd  = a[q][q] - a[p][p];                                                 \
    float den = fabsf(dd) + sqrtf(dd * dd + 4.0f * apq * apq) + 1e-30f;            \
    float tt  = ((dd >= 0.0f) ? (2.0f * apq) : (-2.0f * apq)) / den;               \
    float cc  = 1.0f / sqrtf(1.0f + tt * tt);                                      \
    float ss  = tt * cc;                                                           \
    _Pragma("unroll")                                                              \
    for (int k = 0; k < 4; ++k) {                                                  \
      float u = a[p][k], w = a[q][k];                                              \
      a[p][k] = cc * u - ss * w;                                                   \
      a[q][k] = ss * u + cc * w;                                                   \
    }                                                                              \
    _Pragma("unroll")                                                              \
    for (int k = 0; k < 4; ++k) {                                                  \
      float u = a[k][p], w = a[k][q];                                              \
      a[k][p] = cc * u - ss * w;                                                   \
      a[k][q] = ss * u + cc * w;                                                   \
    }                                                                              \
    _Pragma("unroll")                                                              \
    for (int k = 0; k < 4; ++k) {                                                  \
      float u = vv[k][p], w = vv[k][q];                                            \
      vv[k][p] = cc * u - ss * w;                                                  \
      vv[k][q] = ss * u + cc * w;                                                  \
    }                                                                              \
  } while (0)

__device__ __forceinline__ void jacobi4(float (&a)[4][4], float (&vv)[4][4]) {
#pragma unroll
  for (int i = 0; i < 4; ++i)
#pragma unroll
    for (int j = 0; j < 4; ++j) vv[i][j] = (i == j) ? 1.0f : 0.0f;
  for (int sweep = 0; sweep < 5; ++sweep) {
    JROT(0, 1); JROT(0, 2); JROT(0, 3);
    JROT(1, 2); JROT(1, 3); JROT(2, 3);
  }
}

__device__ __forceinline__ float swap16(float x) {
  // ds_swizzle_b32 SWAPX16: and=0x1f, or=0, xor=0x10  -> lane ^= 16
  return __uint_as_float(__builtin_amdgcn_ds_swizzle(__float_as_uint(x), 0x401F));
}

__device__ __forceinline__ void prefetch_wgp(const void* p) {
  // WGP-scope prefetch: fills all cache levels incl. L0 (SE scope would skip WGP$).
  asm volatile("global_prefetch_b8 %0, off" :: "v"(p));
}

// vec(m1)_b = vec(X_b) . kron(w1,w1)  =>  GEMM M=B, K=1024, N=16 via 16x16x4 f32 WMMA.
// SRC0 (A, 16x4) = kron-table slice (constant, from LDS), SRC1 (B, 4x16) = X rows.
// D = [entry][sample]: vgpr r = (entry r, sample lane) lanes 0-15,
//                      vgpr r = (entry r+8, sample lane-16) lanes 16-31.
__global__ __launch_bounds__(256)
__attribute__((amdgpu_waves_per_eu(4)))
void spdnet_fused(
    const float* __restrict__ x, const float* __restrict__ w1,
    const float* __restrict__ w3, const float* __restrict__ fc,
    float* __restrict__ out, int nB) {
  // 64 KB: kron(w1,w1) pre-swizzled into the WMMA A-operand lane layout:
  // entry t=(s,l): float2 { W(4s+2*(l>>4), l&15), W(4s+2*(l>>4)+1, l&15) },
  // W(k,n) = w1[k>>5][n>>2] * w1[k&31][n&3].
  __shared__ float bws[16384];

  const int tid = threadIdx.x;
  for (int t = tid; t < 8192; t += 256) {
    int s  = t >> 5;
    int l  = t & 31;
    int k0 = 4 * s + ((l >> 4) << 1);
    int n  = l & 15;
    int i4 = n >> 2, j4 = n & 3;
    float b0 = w1[((k0)     >> 5) * 4 + i4] * w1[((k0)     & 31) * 4 + j4];
    float b1 = w1[((k0 + 1) >> 5) * 4 + i4] * w1[((k0 + 1) & 31) * 4 + j4];
    bws[2 * t]     = b0;
    bws[2 * t + 1] = b1;
  }
  __syncthreads();

  const int  lane = tid & 31;
  const int  wv   = tid >> 5;
  const int  hh   = lane >> 4;   // which half-wave
  const int  l16  = lane & 15;
  const long base = ((long)blockIdx.x * 8 + wv) * 32;  // 32 samples per wave
  if (base >= nB) return;

  float m[16];
#pragma unroll
  for (int e = 0; e < 16; ++e) m[e] = 0.0f;

#pragma unroll
  for (int g = 0; g < 2; ++g) {  // two 16-sample WMMA tiles per wave
    // Lane l / l+16 stream the row of sample (base+g*16+l16): {X[4s+2h], X[4s+2h+1]}
    const float* arow = x + (size_t)(base + g * 16 + l16) * 1024 + 2 * hh;
    v8f acc = {0.f, 0.f, 0.f, 0.f, 0.f, 0.f, 0.f, 0.f};

    // Software-pipelined: double-buffered X operands, 8 loads in flight while
    // the dependent WMMA chain consumes the other buffer.
    v2f bufA[8], bufB[8];
#pragma unroll
    for (int i = 0; i < 8; ++i)
      bufA[i] = *reinterpret_cast<const v2f*>(arow + 4 * i);

    for (int so = 0; so < 256; so += 16) {
      if (so < 192) prefetch_wgp(arow + 4 * so + 256);  // 1KB ahead, stays in-row
#pragma unroll
      for (int i = 0; i < 8; ++i)
        bufB[i] = *reinterpret_cast<const v2f*>(arow + 4 * (so + 8 + i));
#pragma unroll
      for (int i = 0; i < 8; ++i) {
        const int s = so + i;
        const v2f A = *reinterpret_cast<const v2f*>(&bws[(s * 32 + lane) * 2]);
        acc = __builtin_amdgcn_wmma_f32_16x16x4_f32(
            false, A, false, bufA[i], (short)0, acc, false, false);
      }
      if (so < 240) {
#pragma unroll
        for (int i = 0; i < 8; ++i)
          bufA[i] = *reinterpret_cast<const v2f*>(arow + 4 * (so + 16 + i));
      }
#pragma unroll
      for (int i = 0; i < 8; ++i) {
        const int s = so + 8 + i;
        const v2f A = *reinterpret_cast<const v2f*>(&bws[(s * 32 + lane) * 2]);
        acc = __builtin_amdgcn_wmma_f32_16x16x4_f32(
            false, A, false, bufB[i], (short)0, acc, false, false);
      }
    }

    // Swap halves so each lane owns one full 16-entry m1 (sample = base + lane).
    float sw[8];
#pragma unroll
    for (int r = 0; r < 8; ++r) sw[r] = swap16(acc[r]);
    if (hh == g) {  // lanes<16 keep tile 0, lanes>=16 keep tile 1
#pragma unroll
      for (int e = 0; e < 8; ++e) {
        m[e]     = hh ? sw[e]  : acc[e];
        m[e + 8] = hh ? acc[e] : sw[e];
      }
    }
  }

  const long smp = base + lane;

  // ---- per-lane SPD tail: ReEig -> bilinear(w3) -> LogEig -> FC ----
  float a[4][4], vv[4][4];
#pragma unroll
  for (int i = 0; i < 4; ++i)
#pragma unroll
    for (int j = 0; j < 4; ++j) a[i][j] = 0.5f * (m[i * 4 + j] + m[j * 4 + i]);

  jacobi4(a, vv);
  float lam[4];
#pragma unroll
  for (int c = 0; c < 4; ++c) lam[c] = fmaxf(a[c][c], REC_EPS);

  float x1m[4][4];
#pragma unroll
  for (int i = 0; i < 4; ++i)
#pragma unroll
    for (int j = 0; j < 4; ++j) {
      float s = 0.0f;
#pragma unroll
      for (int c = 0; c < 4; ++c) s = fmaf(vv[i][c] * lam[c], vv[j][c], s);
      x1m[i][j] = s;
    }

  float w3r[16];
#pragma unroll
  for (int k = 0; k < 16; ++k) w3r[k] = w3[k];  // uniform -> s_loads

  float t3[4][4];
#pragma unroll
  for (int i = 0; i < 4; ++i)
#pragma unroll
    for (int j = 0; j < 4; ++j) {
      float s = 0.0f;
#pragma unroll
      for (int q = 0; q < 4; ++q) s = fmaf(x1m[i][q], w3r[q * 4 + j], s);
      t3[i][j] = s;
    }
#pragma unroll
  for (int i = 0; i < 4; ++i)
#pragma unroll
    for (int j = 0; j < 4; ++j) {
      float s = 0.0f;
#pragma unroll
      for (int p = 0; p < 4; ++p) s = fmaf(w3r[p * 4 + i], t3[p][j], s);
      a[i][j] = s;
    }
#pragma unroll
  for (int i = 0; i < 4; ++i)
#pragma unroll
    for (int j = i + 1; j < 4; ++j) {
      float s = 0.5f * (a[i][j] + a[j][i]);
      a[i][j] = s; a[j][i] = s;
    }

  jacobi4(a, vv);
#pragma unroll
  for (int c = 0; c < 4; ++c) lam[c] = __logf(fmaxf(a[c][c], LOG_FLOOR));

  float feat[16];
#pragma unroll
  for (int i = 0; i < 4; ++i)
#pragma unroll
    for (int j = 0; j < 4; ++j) {
      float s = 0.0f;
#pragma unroll
      for (int c = 0; c < 4; ++c) s = fmaf(vv[i][c] * lam[c], vv[j][c], s);
      feat[i * 4 + j] = s;
    }

  float lg0 = 0.0f, lg1 = 0.0f;
#pragma unroll
  for (int e = 0; e < 16; ++e) {
    lg0 = fmaf(feat[e], fc[2 * e], lg0);
    lg1 = fmaf(feat[e], fc[2 * e + 1], lg1);
  }

  // outputs: [logits B*2][feat B*16]; lane<->sample mapping keeps stores coalesced.
  // Non-temporal: write-once data should not pollute L0/L2 needed by the X stream.
  float* fo = out + (size_t)nB * 2 + (size_t)smp * 16;
  v4f f0 = {feat[0], feat[1], feat[2], feat[3]};
  v4f f1 = {feat[4], feat[5], feat[6], feat[7]};
  v4f f2 = {feat[8], feat[9], feat[10], feat[11]};
  v4f f3 = {feat[12], feat[13], feat[14], feat[15]};
  __builtin_nontemporal_store(f0, reinterpret_cast<v4f*>(fo + 0));
  __builtin_nontemporal_store(f1, reinterpret_cast<v4f*>(fo + 4));
  __builtin_nontemporal_store(f2, reinterpret_cast<v4f*>(fo + 8));
  __builtin_nontemporal_store(f3, reinterpret_cast<v4f*>(fo + 12));
  v2f lg = {lg0, lg1};
  __builtin_nontemporal_store(lg, reinterpret_cast<v2f*>(out + (size_t)smp * 2));
}

extern "C" void kernel_launch(void* const* d_in, const int* in_sizes, int n_in,
                              void* d_out, int out_size, void* d_ws, size_t ws_size,
                              hipStream_t stream) {
  const float* x  = (const float*)d_in[0];
  const float* w1 = (const float*)d_in[1];
  const float* w3 = (const float*)d_in[2];
  const float* fc = (const float*)d_in[3];
  float* out = (float*)d_out;
  int nB = in_sizes[0] / 1024;           // B samples (32x32 each)
  int blocks = (nB + 255) / 256;         // 8 waves/block, 32 samples/wave
  hipLaunchKernelGGL(spdnet_fused, dim3(blocks), dim3(256), 0, stream,
                     x, w1, w3, fc, out, nB);
}